// LocalEnergyHamiltonian_72971494359232
// MI455X (gfx1250) — compile-verified
//
#include <hip/hip_runtime.h>
#include <hip/hip_bf16.h>

typedef __attribute__((ext_vector_type(2))) float v2f;
typedef __attribute__((ext_vector_type(8))) float v8f;

#define BATCH 1024
#define NEL   16
#define DD    48          // NE*3
#define HH    512
#define NATM  4
#define LDA   516         // padded A row stride: 516 mod 64 banks = 4 -> conflict-free frags
#define MIN_DIST 1e-10f

constexpr int SMEM_FLOATS = DD * LDA + 7 * HH + 3 * DD + 16;

__device__ __forceinline__ float wave_reduce_add(float v) {
#pragma unroll
  for (int off = 16; off > 0; off >>= 1) v += __shfl_xor(v, off, 32);
  return v;
}

__global__ __launch_bounds__(256)
void local_energy_kernel(const float* __restrict__ ec, const float* __restrict__ ac,
                         const float* __restrict__ zc, const float* __restrict__ W1,
                         const float* __restrict__ b1v, const float* __restrict__ W2,
                         const float* __restrict__ b2v, const float* __restrict__ w3v,
                         float* __restrict__ out) {
  extern __shared__ float smem[];
  float* As  = smem;                  // [DD][LDA]  (W1 scaled by s1)
  float* h1s = As  + DD * LDA;        // [HH]
  float* s1s = h1s + HH;              // [HH]
  float* v2s = s1s + HH;              // [HH]
  float* cjs = v2s + HH;              // [HH]
  float* v1s = cjs + HH;              // [HH]
  float* uus = v1s + HH;              // [HH]
  float* qqs = uus + HH;              // [HH]
  float* xs  = qqs + HH;              // [DD]
  float* hds = xs  + DD;              // [DD]  hdiag accum (term2 then +term1)
  float* gsh = hds + DD;              // [DD]  gradient
  float* red = gsh + DD;              // [16]  red[0] = potential accum

  const int tid  = threadIdx.x;
  const int lane = tid & 31;
  const int wave = tid >> 5;
  const int bidx = blockIdx.x;
  const float* x = ec + bidx * DD;

  if (tid < DD) xs[tid] = x[tid];
  if (tid == 0) red[0] = 0.0f;
  __syncthreads();

  // ---------------- Potential (cheap, done while hot) ----------------
  if (tid < 120) {                       // e-e pairs i<j  (16*15/2 = 120)
    int i = 0, p = tid, rem = NEL - 1;
    while (p >= rem) { p -= rem; ++i; --rem; }
    int j = i + 1 + p;
    float dx = xs[i*3+0] - xs[j*3+0];
    float dy = xs[i*3+1] - xs[j*3+1];
    float dz = xs[i*3+2] - xs[j*3+2];
    float r = fmaxf(sqrtf(dx*dx + dy*dy + dz*dz), MIN_DIST);
    atomicAdd(&red[0], 1.0f / r);
  } else if (tid < 120 + NEL * NATM) {   // e-n terms
    int t = tid - 120;
    int i = t >> 2, a = t & 3;
    float dx = xs[i*3+0] - ac[a*3+0];
    float dy = xs[i*3+1] - ac[a*3+1];
    float dz = xs[i*3+2] - ac[a*3+2];
    float r = fmaxf(sqrtf(dx*dx + dy*dy + dz*dz), MIN_DIST);
    atomicAdd(&red[0], -zc[a] / r);
  } else if (tid == 255) {               // n-n (constant, tiny)
    float nn = 0.0f;
    for (int i = 0; i < NATM; ++i)
      for (int j = i + 1; j < NATM; ++j) {
        float dx = ac[i*3+0] - ac[j*3+0];
        float dy = ac[i*3+1] - ac[j*3+1];
        float dz = ac[i*3+2] - ac[j*3+2];
        float r = sqrtf(dx*dx + dy*dy + dz*dz);
        if (r > MIN_DIST) nn += zc[i] * zc[j] / r;
      }
    atomicAdd(&red[0], nn);
  }

  // ---------------- Layer 1: a = x W1 + b1 -> h1, s1 ----------------
  for (int i = tid; i < HH; i += 256) {
    float acc = b1v[i];
#pragma unroll
    for (int d = 0; d < DD; ++d) acc += xs[d] * W1[d * HH + i];
    float t = tanhf(acc);
    h1s[i] = t;
    s1s[i] = 1.0f - t * t;
  }
  __syncthreads();

  // ---------------- Stage As = W1 * s1 into LDS ----------------
  for (int idx = tid; idx < DD * HH; idx += 256) {
    int d = idx >> 9, k = idx & (HH - 1);
    As[d * LDA + k] = W1[idx] * s1s[k];
  }

  // ---------------- Layer 2: b = h1 W2 + b2 -> v2, c ----------------
  for (int j = tid; j < HH; j += 256) {
    float acc = b2v[j];
#pragma unroll 4
    for (int i = 0; i < HH; ++i) acc += h1s[i] * W2[i * HH + j];
    float t  = tanhf(acc);
    float s2 = 1.0f - t * t;
    float w  = w3v[j];
    v2s[j] = w * s2;
    cjs[j] = -2.0f * w * t * s2;
  }
  __syncthreads();

  // ---------------- v1 = W2 . v2  (row reduce per wave) ----------------
  for (int i = wave; i < HH; i += 8) {
    float acc = 0.0f;
#pragma unroll 4
    for (int j = lane; j < HH; j += 32) acc += W2[i * HH + j] * v2s[j];
    acc = wave_reduce_add(acc);
    if (lane == 0) v1s[i] = acc;
  }
  __syncthreads();

  for (int i = tid; i < HH; i += 256) {
    float vv = v1s[i];
    uus[i] = vv * s1s[i];
    qqs[i] = -2.0f * h1s[i] * s1s[i] * vv;
  }
  __syncthreads();

  // ---------------- g_d = W1.u ; term2_d = (W1^2).q ----------------
  for (int d = wave * 6; d < wave * 6 + 6; ++d) {
    float ga = 0.0f, ta = 0.0f;
#pragma unroll 4
    for (int i = lane; i < HH; i += 32) {
      float w = W1[d * HH + i];
      ga += w * uus[i];
      ta += w * w * qqs[i];
    }
    ga = wave_reduce_add(ga);
    ta = wave_reduce_add(ta);
    if (lane == 0) { gsh[d] = ga; hds[d] = ta; }
  }
  __syncthreads();

  // ---------------- WMMA: M = As(48x512) @ W2(512x512), term1 ----------------
  // wave w owns N strip [w*64, w*64+64): 3 M-tiles x 4 N-tiles of 16x16.
  const int nb   = wave * 64;
  const int half = lane >> 4;      // 0: lanes 0-15, 1: lanes 16-31
  const int l16  = lane & 15;

  v8f accv[3][4];
#pragma unroll
  for (int mt = 0; mt < 3; ++mt)
#pragma unroll
    for (int nt = 0; nt < 4; ++nt) accv[mt][nt] = v8f{};

  for (int k0 = 0; k0 < HH; k0 += 4) {
    if ((k0 & 63) == 0 && k0 + 64 < HH)
      __builtin_prefetch(&W2[(size_t)(k0 + 64) * HH + nb + l16], 0, 1);

    // A fragments: lane half selects K+0..1 vs K+2..3 (ISA 16x4 f32 layout)
    v2f afrag[3];
#pragma unroll
    for (int mt = 0; mt < 3; ++mt) {
      const float* ap = &As[(mt * 16 + l16) * LDA + k0 + 2 * half];
      afrag[mt] = *(const v2f*)ap;      // ds_read_b64, conflict-free (LDA pad)
    }
#pragma unroll
    for (int nt = 0; nt < 4; ++nt) {
      int n = nb + nt * 16 + l16;
      v2f bfrag;                         // B 4x16: VGPR0 = K+2h, VGPR1 = K+1+2h
      bfrag.x = W2[(size_t)(k0 + 2 * half) * HH + n];
      bfrag.y = W2[(size_t)(k0 + 1 + 2 * half) * HH + n];
#pragma unroll
      for (int mt = 0; mt < 3; ++mt) {
        accv[mt][nt] = __builtin_amdgcn_wmma_f32_16x16x4_f32(
            false, afrag[mt], false, bfrag, (short)0, accv[mt][nt], false, false);
      }
    }
  }

  // term1_d += sum_j c_j * M[d,j]^2 ; C layout: VGPR r -> M = r + 8*half, N = l16
#pragma unroll
  for (int mt = 0; mt < 3; ++mt) {
#pragma unroll
    for (int r = 0; r < 8; ++r) {
      float s = 0.0f;
#pragma unroll
      for (int nt = 0; nt < 4; ++nt) {
        float m = accv[mt][nt][r];
        s += cjs[nb + nt * 16 + l16] * m * m;
      }
#pragma unroll
      for (int off = 1; off < 16; off <<= 1) s += __shfl_xor(s, off, 32);
      if (l16 == 0) atomicAdd(&hds[mt * 16 + r + 8 * half], s);
    }
  }
  __syncthreads();

  // ---------------- Finalize ----------------
  if (wave == 0) {
    float ll = 0.0f, gg = 0.0f;
    for (int d = lane; d < DD; d += 32) {
      float hv = hds[d];
      hv = (fabsf(hv) <= 3.402823466e38f) ? fminf(fmaxf(hv, -50.0f), 50.0f) : 0.0f;
      ll += hv;
      float gv = gsh[d];
      gg += gv * gv;
    }
    ll = wave_reduce_add(ll);
    gg = wave_reduce_add(gg);
    if (lane == 0) {
      float kin = -0.5f * (ll + gg);
      kin = fminf(fmaxf(kin, -100.0f), 100.0f);
      float pot = fminf(fmaxf(red[0], -1000.0f), 1000.0f);
      out[bidx] = kin + pot;
    }
  }
}

extern "C" void kernel_launch(void* const* d_in, const int* in_sizes, int n_in,
                              void* d_out, int out_size, void* d_ws, size_t ws_size,
                              hipStream_t stream) {
  const float* ec = (const float*)d_in[0];   // electron_coords [1024,16,3]
  const float* ac = (const float*)d_in[1];   // atom_coords [4,3]
  const float* zc = (const float*)d_in[2];   // nuclear_charges [4]
  const float* W1 = (const float*)d_in[3];   // [48,512]
  const float* b1 = (const float*)d_in[4];   // [512]
  const float* W2 = (const float*)d_in[5];   // [512,512]
  const float* b2 = (const float*)d_in[6];   // [512]
  const float* w3 = (const float*)d_in[7];   // [512]
  float* out = (float*)d_out;                // [1024]

  size_t smem = (size_t)SMEM_FLOATS * sizeof(float);  // ~114 KB (<< 320 KB WGP LDS)
  hipLaunchKernelGGL(local_energy_kernel, dim3(BATCH), dim3(256), smem, stream,
                     ec, ac, zc, W1, b1, W2, b2, w3, out);
}